// kernel_generated_0_21887153341341
// MI455X (gfx1250) — compile-verified
//
#include <hip/hip_runtime.h>
#include <hip/hip_bf16.h>

typedef __attribute__((ext_vector_type(2))) float v2f;
typedef __attribute__((ext_vector_type(4))) float v4f;
typedef __attribute__((ext_vector_type(8))) float v8f;

#define C_IN   64
#define C_OUT  64
#define HDIM   56
#define WDIM   56
#define KTOT   384      /* C_IN * 6 taps (l=2, k=3) */
#define LROW   72       /* padded LDS row width: 4 left-pad + 56 data + 12 right-pad */

__launch_bounds__(128)
__global__ void shiftconv_wmma_f32(const float* __restrict__ x,
                                   const float* __restrict__ w,
                                   float* __restrict__ y)
{
    __shared__ float xs[2 * C_IN * LROW];   // 36 KB: two shifted x-rows, zero halos

    const int o    = blockIdx.x;            // output row
    const int tid  = threadIdx.x;           // 0..127 (4 waves, wave32)
    const int wv   = tid >> 5;
    const int lane = tid & 31;
    const int m    = lane & 15;             // M (A) / N (B) index within tile
    const int koff = (lane >> 4) << 1;      // 0 for lanes 0-15, 2 for lanes 16-31

    // ---- stage: one (l, j) row per thread, float4-wide, halo-only zeroing ----
    // l=0 tap-group reads x row (o-1)%56 ; l=1 reads (o-2)%56  (cyclic)
    {
        const int l = tid >> 6;
        const int j = tid & 63;
        const int r = (l ? (o + HDIM - 2) : (o + HDIM - 1)) % HDIM;
        float* dst = xs + tid * LROW;                       // 288B-aligned
        const v4f* src = (const v4f*)(x + (j * HDIM + r) * WDIM);  // 224B-aligned
        const v4f z4 = {};
        ((v4f*)dst)[0]  = z4;   // floats 0..3   (left zero pad: covers n+k-1 = -1)
        ((v4f*)dst)[15] = z4;   // floats 60..63 (right pad: n=55,k=2 and n>=56 halo)
        ((v4f*)dst)[16] = z4;   // floats 64..67
        ((v4f*)dst)[17] = z4;   // floats 68..71
        #pragma unroll
        for (int q = 0; q < 14; ++q)
            ((v4f*)(dst + 4))[q] = src[q];                  // data at index n+4
    }
    __syncthreads();

    const int iBase = wv << 4;              // this wave's output-channel tile

    const v8f vzero = {};
    v8f acc[4];
    #pragma unroll
    for (int t = 0; t < 4; ++t) acc[t] = vzero;

    // Effective A matrix: A[i][kk] = W[i*384 + kk], kk = j*6 + l*3 + k
    const float* wrow = w + (size_t)(iBase + m) * KTOT + koff;
    __builtin_prefetch(wrow, 0, 0);

    // kk pattern repeats every 12 (= 3 WMMA steps = 2 in-channels).
    // Precompute 6 per-lane LDS pointers (3 steps x 2 K-slots), decode once.
    // B[kk][n] = x[j, row_l, n+k-1] -> xs[(l*64+j)*LROW + n + k + 3]
    const float* pb[6];
    #pragma unroll
    for (int s = 0; s < 3; ++s) {
        #pragma unroll
        for (int e = 0; e < 2; ++e) {
            const int c  = 4 * s + koff + e;       // kk within period, 0..11
            const int jc = c / 6;
            const int q  = c - jc * 6;
            const int lc = q / 3;
            const int kc = q - lc * 3;
            pb[2 * s + e] = xs + (lc * C_IN + jc) * LROW + kc + 3 + m;
        }
    }

    for (int u = 0; u < KTOT / 12; ++u) {          // 32 periods
        #pragma unroll
        for (int s = 0; s < 3; ++s) {
            const v2f a = *(const v2f*)(wrow + u * 12 + s * 4);  // b64 load
            #pragma unroll
            for (int t = 0; t < 4; ++t) {          // N-tiles: n_base = 0,16,32,48
                v2f b;
                b.x = pb[2 * s + 0][t * 16];       // ds_load_b32, imm offset t*64
                b.y = pb[2 * s + 1][t * 16];
                acc[t] = __builtin_amdgcn_wmma_f32_16x16x4_f32(
                    /*neg_a=*/false, a, /*neg_b=*/false, b,
                    /*c_mod=*/(short)0, acc[t],
                    /*reuse_a=*/false, /*reuse_b=*/false);
            }
        }
        #pragma unroll
        for (int e = 0; e < 6; ++e) pb[e] += 2 * LROW;   // advance 2 in-channels
    }

    // D layout: VGPR v -> M = v (lanes 0-15) / v+8 (lanes 16-31), N = lane%16
    #pragma unroll
    for (int t = 0; t < 4; ++t) {
        const int n = t * 16 + m;
        if (n < WDIM) {
            const float* av = (const float*)&acc[t];
            float* yp = y + (size_t)(iBase + ((lane < 16) ? 0 : 8)) * HDIM * WDIM
                          + o * WDIM + n;
            #pragma unroll
            for (int v = 0; v < 8; ++v)
                yp[(size_t)v * HDIM * WDIM] = av[v];
        }
    }
}

extern "C" void kernel_launch(void* const* d_in, const int* in_sizes, int n_in,
                              void* d_out, int out_size, void* d_ws, size_t ws_size,
                              hipStream_t stream) {
    const float* x = (const float*)d_in[0];   // (1, 64, 56, 56) fp32
    const float* w = (const float*)d_in[1];   // (64, 64, 2, 3)  fp32
    float* y = (float*)d_out;                 // (1, 64, 56, 56) fp32
    (void)in_sizes; (void)n_in; (void)out_size; (void)d_ws; (void)ws_size;
    shiftconv_wmma_f32<<<dim3(HDIM), dim3(128), 0, stream>>>(x, w, y);
}